// MessagePassingLayer_352187319219
// MI455X (gfx1250) — compile-verified
//
#include <hip/hip_runtime.h>
#include <hip/hip_bf16.h>

#define NUM_NODES 40000
#define DIM       128

typedef __attribute__((ext_vector_type(2))) float v2f;
typedef __attribute__((ext_vector_type(8))) float v8f;

// ---------------------------------------------------------------------------
// Kernel 1: zero the workspace (agg [NUM_NODES*DIM] floats, then deg [NUM_NODES])
// ---------------------------------------------------------------------------
__global__ void mp_zero_ws(float* __restrict__ p, size_t n) {
    size_t i = (size_t)blockIdx.x * blockDim.x + threadIdx.x;
    if (i < n) p[i] = 0.0f;
}

// ---------------------------------------------------------------------------
// Kernel 2: edge scatter-aggregate.  One wave32 per edge: each lane handles 4
// consecutive features (float4 load -> 4 global_atomic_add_f32).  Lane 0 also
// bumps the degree counter.  x and agg both fit in the 192 MB L2, so the
// random gather + atomic traffic stays on-chip.
// ---------------------------------------------------------------------------
__global__ void mp_scatter_edges(const float* __restrict__ x,
                                 const int*   __restrict__ ei,   // [2, E] int32
                                 float* __restrict__ agg,
                                 float* __restrict__ deg,
                                 int E) {
    int gid  = blockIdx.x * blockDim.x + threadIdx.x;
    int e    = gid >> 5;
    int lane = gid & 31;
    if (e >= E) return;

    int src = ei[e];
    int dst = ei[E + e];

    const float4 v = *(const float4*)(x + (size_t)src * DIM + lane * 4);
    float* d = agg + (size_t)dst * DIM + lane * 4;

    __hip_atomic_fetch_add(d + 0, v.x, __ATOMIC_RELAXED, __HIP_MEMORY_SCOPE_AGENT);
    __hip_atomic_fetch_add(d + 1, v.y, __ATOMIC_RELAXED, __HIP_MEMORY_SCOPE_AGENT);
    __hip_atomic_fetch_add(d + 2, v.z, __ATOMIC_RELAXED, __HIP_MEMORY_SCOPE_AGENT);
    __hip_atomic_fetch_add(d + 3, v.w, __ATOMIC_RELAXED, __HIP_MEMORY_SCOPE_AGENT);
    if (lane == 0)
        __hip_atomic_fetch_add(deg + dst, 1.0f, __ATOMIC_RELAXED, __HIP_MEMORY_SCOPE_AGENT);
}

// ---------------------------------------------------------------------------
// Kernel 3: out = (agg / max(deg,1)) @ W^T + (deg>0 ? b : 0)
// One wave32 per 16(M)x16(N) tile, K=128 swept in 32 steps of
// V_WMMA_F32_16X16X4_F32 (full fp32 precision; the GEMM is bandwidth-bound so
// there is no reason to downcast).
//
// Per-lane VGPR layouts (ISA 7.12.2), with half = lane>>4, nl = lane&15:
//   A (16x4):  row M = nl,           a = { A[row][k0+2*half], A[row][k0+2*half+1] }
//   B (4x16):  col N = nl,           b = { Bt[k0+2*half][col], Bt[k0+2*half+1][col] }
//   C/D (16x16, 8 VGPRs): c[r] is element (M = r + 8*half, N = nl)
// Tile selection is wave-uniform, so EXEC is all-ones at every WMMA.
// ---------------------------------------------------------------------------
__global__ void mp_gemm_wmma(const float* __restrict__ agg,
                             const float* __restrict__ deg,
                             const float* __restrict__ W,     // [DIM out][DIM in] row-major
                             const float* __restrict__ bias,  // [DIM]
                             float* __restrict__ out,         // [M][DIM]
                             int M) {
    const int tilesN = DIM / 16;                                   // 8
    const int wave   = (blockIdx.x * blockDim.x + threadIdx.x) >> 5;
    const int lane   = threadIdx.x & 31;
    const int tm     = wave / tilesN;
    const int tn     = wave % tilesN;
    if (tm * 16 >= M) return;                                      // wave-uniform

    const int half = lane >> 4;
    const int nl   = lane & 15;

    const int rowA = tm * 16 + nl;          // A row owned by this lane
    const int colB = tn * 16 + nl;          // B column owned by this lane

    const float inv  = 1.0f / fmaxf(deg[rowA], 1.0f);
    const float* arow = agg + (size_t)rowA * DIM;
    const float* wrow = W   + (size_t)colB * DIM;   // W^T[k][colB] == W[colB][k]

    v8f c = {};
    #pragma unroll
    for (int k0 = 0; k0 < DIM; k0 += 4) {
        const int ka = k0 + 2 * half;
        v2f a, b;
        a.x = arow[ka]     * inv;
        a.y = arow[ka + 1] * inv;
        b.x = wrow[ka];
        b.y = wrow[ka + 1];
        // 8 args: (neg_a, A, neg_b, B, c_mod, C, reuse_a, reuse_b)
        c = __builtin_amdgcn_wmma_f32_16x16x4_f32(
                /*neg_a=*/false, a, /*neg_b=*/false, b,
                /*c_mod=*/(short)0, c, /*reuse_a=*/false, /*reuse_b=*/false);
    }

    // Store D tile with fused conditional bias.
    const int col = tn * 16 + nl;
    const float bcol = bias[col];
    #pragma unroll
    for (int r = 0; r < 8; ++r) {
        const int row = tm * 16 + r + 8 * half;
        const float bb = (deg[row] > 0.0f) ? bcol : 0.0f;
        out[(size_t)row * DIM + col] = c[r] + bb;
    }
}

// ---------------------------------------------------------------------------
// Launcher.  Inputs (setup_inputs order): x [N*DIM] f32, edge_index [2*E] int,
// W [DIM*DIM] f32, b [DIM] f32.  Workspace: agg (N*DIM f32) | deg (N f32).
// ---------------------------------------------------------------------------
extern "C" void kernel_launch(void* const* d_in, const int* in_sizes, int n_in,
                              void* d_out, int out_size, void* d_ws, size_t ws_size,
                              hipStream_t stream) {
    const float* x  = (const float*)d_in[0];
    const int*   ei = (const int*)  d_in[1];
    const float* W  = (const float*)d_in[2];
    const float* b  = (const float*)d_in[3];
    float* out = (float*)d_out;

    const int Nn = in_sizes[0] / DIM;       // 40000
    const int E  = in_sizes[1] / 2;         // 640000

    float* agg = (float*)d_ws;
    float* deg = agg + (size_t)Nn * DIM;

    // 1) zero agg + deg
    size_t zn = (size_t)Nn * DIM + (size_t)Nn;
    mp_zero_ws<<<(unsigned)((zn + 255) / 256), 256, 0, stream>>>(agg, zn);

    // 2) scatter-aggregate over edges (one wave32 per edge)
    size_t sthreads = (size_t)E * 32;
    mp_scatter_edges<<<(unsigned)((sthreads + 255) / 256), 256, 0, stream>>>(
        x, ei, agg, deg, E);

    // 3) fused mean + GEMM + bias via WMMA (8 waves / 256-thread block)
    int tiles = ((Nn + 15) / 16) * (DIM / 16);          // 20000 waves
    mp_gemm_wmma<<<(tiles + 7) / 8, 256, 0, stream>>>(agg, deg, W, b, out, Nn);
}